// GCNLayer_7997229105213
// MI455X (gfx1250) — compile-verified
//
#include <hip/hip_runtime.h>
#include <math.h>
#include <stdint.h>

// Problem constants (match reference)
constexpr int N_USERS = 100000;
constexpr int N_ITEMS = 100000;
constexpr int N_NODES = N_USERS + N_ITEMS;   // 200000
constexpr int D       = 128;                 // feature dim

// ---------------------------------------------------------------------------
// Zero a region with 128-bit stores (global_store_b128).
// n4 = number of uint4 elements (region must be 16B-divisible; ours are).
// ---------------------------------------------------------------------------
__global__ void zero_b128(uint4* __restrict__ p, long long n4) {
    long long i = (long long)blockIdx.x * blockDim.x + threadIdx.x;
    const long long stride = (long long)gridDim.x * blockDim.x;
    const uint4 z = {0u, 0u, 0u, 0u};
    for (; i < n4; i += stride) p[i] = z;
}

// ---------------------------------------------------------------------------
// Degree histogram: out_deg[src[e]]++, in_deg[dst[e]]++ (discarded results ->
// non-returning global_atomic_add_u32, STOREcnt-tracked, resolved at L2).
// ---------------------------------------------------------------------------
__global__ void degree_kernel(const int* __restrict__ src,
                              const int* __restrict__ dst,
                              uint32_t* __restrict__ out_deg,
                              uint32_t* __restrict__ in_deg,
                              int E) {
    int i = blockIdx.x * blockDim.x + threadIdx.x;
    const int stride = gridDim.x * blockDim.x;
    for (; i < E; i += stride) {
        atomicAdd(&out_deg[src[i]], 1u);
        atomicAdd(&in_deg[dst[i]], 1u);
    }
}

// ---------------------------------------------------------------------------
// Edge scatter: persistent wave32s, grid-stride over edges.
//  - Lane l owns columns {l, l+32, l+64, l+96}: every load and every f32
//    atomic-add the wave issues covers one contiguous 128 B span (perfectly
//    coalesced on wave32).
//  - out (102 MB) fits in the 192 MB global L2, so the non-returning
//    global_atomic_add_f32 ops accumulate at the L2 atomic units.
//  - Next iteration's 512 B source row is prefetched (global_prefetch_b8,
//    lanes 16 B apart cover the row) to overlap the random gather latency
//    with the current edge's atomics.
// ---------------------------------------------------------------------------
__global__ __launch_bounds__(256) void scatter_kernel(
        const float* __restrict__ u_f,
        const float* __restrict__ v_f,
        const int* __restrict__ src,
        const int* __restrict__ dst,
        const uint32_t* __restrict__ out_deg,
        float* __restrict__ out,
        int E) {
    const int lane = threadIdx.x & 31;
    const int wavesPerBlock = blockDim.x >> 5;
    const long long wstride = (long long)gridDim.x * wavesPerBlock;
    long long w = (long long)blockIdx.x * wavesPerBlock + (threadIdx.x >> 5);

    for (; w < E; w += wstride) {
        // Wave-uniform endpoints -> SGPRs (scalar row-base math, SMEM deg load).
        const int s = __builtin_amdgcn_readfirstlane(src[w]);
        const int t = __builtin_amdgcn_readfirstlane(dst[w]);

        // node_f = concat(u_f, v_f) without materializing it.
        const float* __restrict__ row =
            (s < N_USERS) ? (u_f + (long long)s * D)
                          : (v_f + (long long)(s - N_USERS) * D);

        // Software prefetch of the next edge's source row.
        const long long wn = w + wstride;
        if (wn < E) {
            const int sn = src[wn];
            const float* rown =
                (sn < N_USERS) ? (u_f + (long long)sn * D)
                               : (v_f + (long long)(sn - N_USERS) * D);
            __builtin_prefetch(rown + (lane << 2), 0, 0);
        }

        // out-degree pre-normalization: clip(deg,1)^-0.5
        const uint32_t dg = out_deg[s];
        const float scale = 1.0f / sqrtf((float)(dg < 1u ? 1u : dg));

        float* __restrict__ orow = out + (long long)t * D;

        const float x0 = row[lane      ];
        const float x1 = row[lane + 32 ];
        const float x2 = row[lane + 64 ];
        const float x3 = row[lane + 96 ];

        // Non-returning f32 atomics (result discarded).
        atomicAdd(&orow[lane      ], x0 * scale);
        atomicAdd(&orow[lane + 32 ], x1 * scale);
        atomicAdd(&orow[lane + 64 ], x2 * scale);
        atomicAdd(&orow[lane + 96 ], x3 * scale);
    }
}

// ---------------------------------------------------------------------------
// Post-aggregation in-degree normalization, 128-bit vectorized streaming:
// out[i] *= clip(in_deg[row],1)^-0.5. 32 float4 per row -> node = i4 >> 5.
// ---------------------------------------------------------------------------
__global__ void norm_kernel(float4* __restrict__ out4,
                            const uint32_t* __restrict__ in_deg,
                            long long n4) {
    long long i = (long long)blockIdx.x * blockDim.x + threadIdx.x;
    const long long stride = (long long)gridDim.x * blockDim.x;
    for (; i < n4; i += stride) {
        const int node = (int)(i >> 5);          // (i*4)/128
        const uint32_t dg = in_deg[node];
        const float s = 1.0f / sqrtf((float)(dg < 1u ? 1u : dg));
        float4 v = out4[i];
        v.x *= s; v.y *= s; v.z *= s; v.w *= s;
        out4[i] = v;
    }
}

// ---------------------------------------------------------------------------
// Host launcher. Inputs (setup_inputs order):
//   d_in[0] = u_f  float32 [N_USERS * D]
//   d_in[1] = v_f  float32 [N_ITEMS * D]
//   d_in[2] = src  int32   [E]
//   d_in[3] = dst  int32   [E]
// d_out = float32 [N_NODES * D]
// d_ws:  [0, N)  out_deg (u32), [N, 2N) in_deg (u32)  -> 1.6 MB needed.
// ---------------------------------------------------------------------------
extern "C" void kernel_launch(void* const* d_in, const int* in_sizes, int n_in,
                              void* d_out, int out_size, void* d_ws, size_t ws_size,
                              hipStream_t stream) {
    const float* u_f = (const float*)d_in[0];
    const float* v_f = (const float*)d_in[1];
    const int*   src = (const int*)d_in[2];
    const int*   dst = (const int*)d_in[3];
    const int    E   = in_sizes[2];

    float*    out     = (float*)d_out;
    uint32_t* out_deg = (uint32_t*)d_ws;
    uint32_t* in_deg  = out_deg + N_NODES;

    // 1) Zero degree arrays (2N u32 = 100000 uint4) and output (N*D f32).
    {
        const long long n4 = (2LL * N_NODES) / 4;
        const int blocks = (int)((n4 + 255) / 256);
        zero_b128<<<blocks, 256, 0, stream>>>((uint4*)d_ws, n4);
    }
    {
        const long long n4 = ((long long)N_NODES * D) / 4;   // 6.4M
        const int blocks = (int)((n4 + 255) / 256);
        zero_b128<<<blocks, 256, 0, stream>>>((uint4*)d_out, n4);
    }

    // 2) Degree histograms.
    {
        const int blocks = (E + 255) / 256;
        degree_kernel<<<blocks, 256, 0, stream>>>(src, dst, out_deg, in_deg, E);
    }

    // 3) Edge scatter: persistent waves, ~10 edges per wave.
    {
        const int blocks = 40000;   // 320K waves for 3.2M edges
        scatter_kernel<<<blocks, 256, 0, stream>>>(u_f, v_f, src, dst,
                                                   out_deg, out, E);
    }

    // 4) In-degree normalization (streaming, b128).
    {
        const long long n4 = ((long long)N_NODES * D) / 4;
        const int blocks = (int)((n4 + 255) / 256);
        norm_kernel<<<blocks, 256, 0, stream>>>((float4*)out, in_deg, n4);
    }
}